// DrugBAN_61203283968295
// MI455X (gfx1250) — compile-verified
//
#include <hip/hip_runtime.h>

// ---------------------------------------------------------------------------
// DrugBAN head, algebraically reduced:
//   softmax rows sum to 1  =>  drug_ctx = segment_sum(drug_x)/NT
//                              tgt_ctx  = mean_t(target_h)
// Both reductions are done on the CDNA5 matrix pipe via the exact-fp32
// V_WMMA_F32_16X16X4_F32 with an all-ones A matrix:
//   D[m,n] = sum_k 1 * B[k,n]   (reduces 4 rows x 16 cols per instruction)
// Result row M=0 lives in c[0] on lanes 0..15 (N = lane).
// ---------------------------------------------------------------------------

typedef float v2f __attribute__((ext_vector_type(2)));
typedef float v8f __attribute__((ext_vector_type(8)));

#define BATCH   64
#define NT      1024
#define D_DIM   256
#define E_DIM   256
#define NMAX    512
#define OUT_DIM (D_DIM + E_DIM)

// ---------------------------------------------------------------------------
// Kernel 1: segment boundaries. batch_idx is sorted, so starts[b] = first i
// with batch_idx[i] >= b (lower bound). starts[64] = N. Handles empty
// segments naturally. Deterministic integer work.
// ---------------------------------------------------------------------------
__global__ void seg_starts_kernel(const int* __restrict__ batch_idx, int n,
                                  int* __restrict__ starts) {
    int b = blockIdx.x * blockDim.x + threadIdx.x;
    if (b > BATCH) return;
    int lo = 0, hi = n;
    while (lo < hi) {
        int mid = (lo + hi) >> 1;
        if (batch_idx[mid] < b) lo = mid + 1; else hi = mid;
    }
    starts[b] = lo;
}

// ---------------------------------------------------------------------------
// Kernel 2: tgt_ctx[b, e] = (1/NT) * sum_t target_h[b, t, e]
// One wave per (e-chunk of 16, batch). Each WMMA folds 4 t-rows; two
// independent accumulators hide the WMMA D->C RAW latency.
// ---------------------------------------------------------------------------
__global__ void __launch_bounds__(32)
tgt_mean_kernel(const float* __restrict__ target_h, float* __restrict__ out) {
    const int lane = threadIdx.x;      // 0..31 (wave32)
    const int ec   = blockIdx.x;       // e-chunk: 16 columns each
    const int b    = blockIdx.y;       // batch
    const int n    = lane & 15;        // N column within chunk
    const int kh   = lane >> 4;        // lane-half -> which pair of K rows

    const float* base = target_h + ((size_t)b * NT) * E_DIM + ec * 16 + n;

    const v2f ones = {1.0f, 1.0f};     // A matrix: all ones (layout-agnostic)
    v8f c0 = {}, c1 = {};

    for (int t = 0; t < NT; t += 8) {
        // prefetch ~4 iterations ahead (emits global_prefetch_b8; OOB
        // speculative prefetches are silently dropped)
        __builtin_prefetch(base + (size_t)(t + 32) * E_DIM, 0, 1);

        const float* p0 = base + (size_t)(t + 0 + kh * 2) * E_DIM;
        const float* p1 = base + (size_t)(t + 4 + kh * 2) * E_DIM;
        v2f bv0, bv1;
        bv0.x = p0[0];
        bv0.y = p0[E_DIM];
        bv1.x = p1[0];
        bv1.y = p1[E_DIM];
        c0 = __builtin_amdgcn_wmma_f32_16x16x4_f32(false, ones, false, bv0,
                                                   (short)0, c0, false, false);
        c1 = __builtin_amdgcn_wmma_f32_16x16x4_f32(false, ones, false, bv1,
                                                   (short)0, c1, false, false);
    }

    if (lane < 16) {
        float s = c0[0] + c1[0];       // row M=0 holds the column sums
        out[(size_t)b * OUT_DIM + D_DIM + ec * 16 + n] = s * (1.0f / NT);
    }
}

// ---------------------------------------------------------------------------
// Kernel 3: drug_ctx[b, d] = (1/NT) * sum over first min(count, NMAX) rows of
// segment b of drug_x[:, d]. Ragged tail masked per-lane with selects (EXEC
// stays all-ones around the WMMA, as the ISA requires).
// ---------------------------------------------------------------------------
__global__ void __launch_bounds__(32)
drug_sum_kernel(const float* __restrict__ drug_x,
                const int* __restrict__ starts, float* __restrict__ out) {
    const int lane = threadIdx.x;
    const int ec   = blockIdx.x;
    const int b    = blockIdx.y;
    const int n    = lane & 15;
    const int kh   = lane >> 4;

    int s   = starts[b];
    int cnt = starts[b + 1] - s;
    if (cnt > NMAX) cnt = NMAX;        // reference pads with mode="drop"
    const int endr = s + cnt;

    const float* base = drug_x + ec * 16 + n;

    const v2f ones = {1.0f, 1.0f};
    v8f c = {};

    for (int r = s; r < endr; r += 4) { // uniform trip count across the wave
        int r0 = r + kh * 2;
        int r1 = r0 + 1;
        v2f bv;
        bv.x = (r0 < endr) ? base[(size_t)r0 * D_DIM] : 0.0f;
        bv.y = (r1 < endr) ? base[(size_t)r1 * D_DIM] : 0.0f;
        c = __builtin_amdgcn_wmma_f32_16x16x4_f32(false, ones, false, bv,
                                                  (short)0, c, false, false);
    }

    if (lane < 16)
        out[(size_t)b * OUT_DIM + ec * 16 + n] = c[0] * (1.0f / NT);
}

// ---------------------------------------------------------------------------
extern "C" void kernel_launch(void* const* d_in, const int* in_sizes, int n_in,
                              void* d_out, int out_size, void* d_ws, size_t ws_size,
                              hipStream_t stream) {
    const float* drug_x    = (const float*)d_in[0];
    const int*   batch_idx = (const int*)d_in[1];
    const float* target_h  = (const float*)d_in[2];
    // d_in[3] = W_a: cancels algebraically (softmax row sums are exactly 1).
    float* out = (float*)d_out;

    const int n_nodes = in_sizes[1];   // 16384
    int* starts = (int*)d_ws;          // 65 ints of scratch

    seg_starts_kernel<<<1, 128, 0, stream>>>(batch_idx, n_nodes, starts);

    dim3 grid(E_DIM / 16, BATCH);      // (16 e-chunks, 64 batches)
    tgt_mean_kernel<<<grid, 32, 0, stream>>>(target_h, out);
    drug_sum_kernel<<<grid, 32, 0, stream>>>(drug_x, starts, out);
}